// RPN_to_RoI_82343112999672
// MI455X (gfx1250) — compile-verified
//
#include <hip/hip_runtime.h>

#define BB 8
#define HH 96
#define WWD 96
#define AA 9
#define NN (HH*WWD*AA)        // 82944
#define NSQ 131072            // next pow2 >= NN
#define KPRE 4000
#define MAXOUT 2000
#define IOU_THR 0.7f
#define SCORE_THR 0.5f
#define NEGV (-1000000000.0f)
#define NW 126                // uint32 words per mask row (4032 bits >= 4000)
#define CH 8192               // LDS sort chunk (64 KB of u64 keys)

typedef float v2f __attribute__((ext_vector_type(2)));
typedef float v8f __attribute__((ext_vector_type(8)));
typedef unsigned long long u64;

__device__ __forceinline__ unsigned orderKey(float f) {
  unsigned s = __float_as_uint(f);
  return (s & 0x80000000u) ? ~s : (s | 0x80000000u);
}

// keys[b][n] = (orderedScore << 32) | ~n  (descending sort => score desc, index asc on ties)
__global__ void init_keys(const float* __restrict__ score, u64* __restrict__ keys) {
  int gid = blockIdx.x * blockDim.x + threadIdx.x;  // BB*NSQ threads
  int b = gid >> 17;
  int n = gid & (NSQ - 1);
  u64 key = 0ull;
  if (n < NN) {
    float sc = score[b * NN + n];
    float m = (sc > SCORE_THR) ? sc : NEGV;
    key = ((u64)orderKey(m) << 32) | (unsigned)(~(unsigned)n);
  }
  keys[gid] = key;
}

// Global comparator step (only used for j >= CH)
__global__ void bitonic_step(u64* __restrict__ keys, int j, int k) {
  int gid = blockIdx.x * blockDim.x + threadIdx.x;
  int b = gid >> 17;
  int i = gid & (NSQ - 1);
  int ixj = i ^ j;
  if (ixj > i) {
    u64* kb = keys + ((size_t)b << 17);
    u64 a = kb[i];
    u64 c = kb[ixj];
    bool up = ((i & k) == 0);
    if ((a < c) == up) { kb[i] = c; kb[ixj] = a; }  // descending sort
  }
}

// All bitonic phases with j < CH run inside LDS (320 KB/WGP makes 64 KB chunks cheap).
// kLow..kHigh: run phases k2 = kLow, 2*kLow, ..., kHigh; for each, j2 from
// min(k2/2, CH/2) down to 1. Direction uses the within-batch index, so this is
// step-for-step identical to the global network.
__global__ void bitonic_lds(u64* __restrict__ keys, int kLow, int kHigh) {
  extern __shared__ u64 lds[];
  size_t base = (size_t)blockIdx.x * CH;     // chunks never straddle a batch (NSQ % CH == 0)
  u64* g = keys + base;
  for (int t = threadIdx.x; t < CH; t += blockDim.x) lds[t] = g[t];
  __syncthreads();
  for (int k2 = kLow; k2 <= kHigh; k2 <<= 1) {
    int j0 = ((k2 >> 1) < (CH >> 1)) ? (k2 >> 1) : (CH >> 1);
    for (int j2 = j0; j2 > 0; j2 >>= 1) {
      for (int p = threadIdx.x; p < (CH >> 1); p += blockDim.x) {
        int i = ((p & ~(j2 - 1)) << 1) | (p & (j2 - 1));
        int ix = i | j2;
        u64 a = lds[i], c2 = lds[ix];
        int ibat = (int)((base + (size_t)i) & (size_t)(NSQ - 1));
        bool up = ((ibat & k2) == 0);
        if ((a < c2) == up) { lds[i] = c2; lds[ix] = a; }
      }
      __syncthreads();
    }
  }
  for (int t = threadIdx.x; t < CH; t += blockDim.x) g[t] = lds[t];
}

// Top-4000 per batch: decode candidate boxes + validity flag
__global__ void gather_decode(const u64* __restrict__ keys,
                              const float* __restrict__ delta,
                              const float* __restrict__ anchors,
                              float4* __restrict__ cand,
                              float* __restrict__ valid) {
  int gid = blockIdx.x * blockDim.x + threadIdx.x;
  if (gid >= BB * KPRE) return;
  int b = gid / KPRE;
  int r = gid - b * KPRE;
  u64 key = keys[((size_t)b << 17) + r];
  unsigned n = ~(unsigned)key;
  unsigned u = (unsigned)(key >> 32);
  unsigned bits = (u & 0x80000000u) ? (u & 0x7FFFFFFFu) : ~u;
  float mval = __uint_as_float(bits);
  valid[gid] = (mval > NEGV * 0.5f) ? 1.0f : 0.0f;
  const float* d = delta + ((size_t)b * NN + n) * 4;
  const float* a = anchors + (size_t)n * 4;
  float tx = d[0], ty = d[1], tw = d[2], th = d[3];
  float a0 = a[0], a1 = a[1], a2 = a[2], a3 = a[3];
  float xa = (a0 + a1) * 0.5f, ya = (a2 + a3) * 0.5f;
  float wa = a1 - a0, ha = a3 - a2;
  float x = tx * wa + xa;
  float y = ty * ha + ya;
  float w = expf(tw) * wa;
  float h = expf(th) * ha;
  float4 bx;
  bx.x = fminf(fmaxf(x - w * 0.5f, 0.f), 1.f);
  bx.y = fminf(fmaxf(x + w * 0.5f, 0.f), 1.f);
  bx.z = fminf(fmaxf(y - h * 0.5f, 0.f), 1.f);
  bx.w = fminf(fmaxf(y + h * 0.5f, 0.f), 1.f);
  cand[gid] = bx;
}

// One wave per 16x16 tile. area_i + area_j outer-sum on the matrix pipe
// (two V_WMMA_F32_16X16X4_F32 rank-1 accumulations), min/max intersection on VALU,
// suppression bits via ballot -> atomicOr. IoU path uses _rn intrinsics so each
// intermediate is individually rounded (no FMA contraction), like the reference.
__global__ void iou_mask(const float4* __restrict__ cand,
                         unsigned* __restrict__ mask) {
  int b = blockIdx.y;
  int it = blockIdx.x >> 5;        // 0..249
  int jtg = blockIdx.x & 31;
  int wv = threadIdx.x >> 5;       // 8 waves per block
  int lane = threadIdx.x & 31;
  int jt = jtg * 8 + wv;           // wave-uniform
  if (jt >= 250) return;           // wave-uniform exit: EXEC stays all-ones
  int i0 = it * 16, j0 = jt * 16;
  const float4* cb = cand + (size_t)b * KPRE;
  float4 bj = cb[j0 + (lane & 15)];
  float area_j = __fmul_rn(__fsub_rn(bj.y, bj.x), __fsub_rn(bj.w, bj.z));
  float4 bii = cb[i0 + (lane & 15)];
  float area_i = __fmul_rn(__fsub_rn(bii.y, bii.x), __fsub_rn(bii.w, bii.z));
  bool lo = lane < 16;
  v2f a1; a1.x = lo ? area_i : 0.f; a1.y = 0.f;   // A[m][0] = area_i[m]
  v2f b1; b1.x = lo ? 1.f : 0.f;    b1.y = 0.f;   // B[0][n] = 1
  v2f a2; a2.x = lo ? 1.f : 0.f;    a2.y = 0.f;   // A[m][0] = 1
  v2f b2; b2.x = lo ? area_j : 0.f; b2.y = 0.f;   // B[0][n] = area_j[n]
  v8f csum = {};
  csum = __builtin_amdgcn_wmma_f32_16x16x4_f32(false, a1, false, b1, (short)0, csum, false, false);
  csum = __builtin_amdgcn_wmma_f32_16x16x4_f32(false, a2, false, b2, (short)0, csum, false, false);
  // C/D layout: lane L, vgpr r -> (M = r + 8*(L>=16), N = L&15)
  int mbase = i0 + ((lane >> 4) << 3);
  unsigned wword = (unsigned)(j0 >> 5);
  unsigned shift = (unsigned)(j0 & 31);            // 0 or 16
  unsigned* mb = mask + (size_t)b * KPRE * NW;
#pragma unroll
  for (int r = 0; r < 8; ++r) {
    float4 bi = cb[mbase + r];
    float iw = fmaxf(__fsub_rn(fminf(bi.y, bj.y), fmaxf(bi.x, bj.x)), 0.f);
    float ih = fmaxf(__fsub_rn(fminf(bi.w, bj.w), fmaxf(bi.z, bj.z)), 0.f);
    float inter = __fmul_rn(iw, ih);
    float uni = __fsub_rn(csum[r], inter);         // area_i + area_j - inter
    float iou = __fdiv_rn(inter, fmaxf(uni, 1e-8f));
    unsigned long long bal = __ballot(iou > IOU_THR);
    if (lane == 0) {
      unsigned blo = (unsigned)bal & 0xFFFFu;           // rows i0+r,  cols j0..j0+15
      unsigned bhi = ((unsigned)(bal >> 16)) & 0xFFFFu; // rows i0+r+8
      atomicOr(mb + (size_t)(i0 + r) * NW + wword, blo << shift);
      atomicOr(mb + (size_t)(i0 + r + 8) * NW + wword, bhi << shift);
    }
  }
}

// Greedy reduction: one block per batch, LDS "removed" bitset.
__global__ void nms_serial(const float4* __restrict__ cand,
                           const float* __restrict__ valid,
                           const unsigned* __restrict__ mask,
                           float4* __restrict__ out) {
  int b = blockIdx.x;
  __shared__ unsigned removed[NW];
  for (int w = threadIdx.x; w < NW; w += blockDim.x) removed[w] = 0u;
  __syncthreads();
  const unsigned* mb = mask + (size_t)b * KPRE * NW;
  const float4* cb = cand + (size_t)b * KPRE;
  const float* vb = valid + (size_t)b * KPRE;
  int count = 0;
  for (int i = 0; i < KPRE; ++i) {
    if (vb[i] == 0.0f) break;                         // uniform: sorted, tail is all invalid
    bool rem = ((removed[i >> 5] >> (i & 31)) & 1u) != 0u;  // uniform
    if (!rem) {
      if (threadIdx.x == 0) out[(size_t)b * MAXOUT + count] = cb[i];
      const unsigned* row = mb + (size_t)i * NW;
      for (int w = threadIdx.x; w < NW; w += blockDim.x) removed[w] |= row[w];
      ++count;
      if (count >= MAXOUT) break;                     // uniform
      __syncthreads();
    }
  }
}

extern "C" void kernel_launch(void* const* d_in, const int* in_sizes, int n_in,
                              void* d_out, int out_size, void* d_ws, size_t ws_size,
                              hipStream_t stream) {
  (void)in_sizes; (void)n_in; (void)ws_size;
  const float* score   = (const float*)d_in[0];
  const float* delta   = (const float*)d_in[1];
  const float* anchors = (const float*)d_in[2];
  char* ws = (char*)d_ws;
  u64*      keys  = (u64*)(ws);                       // 8*131072*8  = 8,388,608 B
  float4*   cand  = (float4*)(ws + 8388608);          // 8*4000*16   =   512,000 B
  float*    valid = (float*)(ws + 8900608);           // 8*4000*4    =   128,000 B
  unsigned* mask  = (unsigned*)(ws + 9028608);        // 8*4000*126*4= 16,128,000 B

  const int nChunks = BB * (NSQ / CH);                // 128 blocks
  const size_t ldsBytes = (size_t)CH * sizeof(u64);   // 64 KB dynamic LDS

  init_keys<<<(BB * NSQ) / 256, 256, 0, stream>>>(score, keys);
  // Phases k = 2..CH entirely in LDS:
  bitonic_lds<<<nChunks, 1024, ldsBytes, stream>>>(keys, 2, CH);
  // Phases k = 2*CH..NSQ: global steps for j >= CH, then LDS tail (j < CH):
  for (int k = CH << 1; k <= NSQ; k <<= 1) {
    for (int j = k >> 1; j >= CH; j >>= 1)
      bitonic_step<<<(BB * NSQ) / 256, 256, 0, stream>>>(keys, j, k);
    bitonic_lds<<<nChunks, 1024, ldsBytes, stream>>>(keys, k, k);
  }
  gather_decode<<<(BB * KPRE + 255) / 256, 256, 0, stream>>>(keys, delta, anchors, cand, valid);
  hipMemsetAsync(mask, 0, (size_t)BB * KPRE * NW * 4, stream);
  iou_mask<<<dim3(250 * 32, BB), 256, 0, stream>>>(cand, mask);
  hipMemsetAsync(d_out, 0, (size_t)out_size * 4, stream);
  nms_serial<<<BB, 128, 0, stream>>>(cand, valid, mask, (float4*)d_out);
}